// LapDeform_28441273434703
// MI455X (gfx1250) — compile-verified
//
#include <hip/hip_runtime.h>
#include <math.h>
#include <stdint.h>

#define KNN 16
#define BLOCK 128
#define WAVES_PER_BLOCK (BLOCK / 32)
// per-wave LDS dwords: KNN neighbors * 32 lanes * 3 dwords * 2 arrays (new, old)
#define WAVE_LDS_DW (KNN * 32 * 3 * 2)
#define OLD_OFF_DW (KNN * 32 * 3)

// Issue an async gather of 12 contiguous bytes (one xyz row) from global to LDS.
// INST_OFFSET is added to BOTH the global and the LDS address (ISA 08 §4.4),
// so one lds/global address pair covers all three dwords via offset:0/4/8.
__device__ __forceinline__ void async_gather12(uint32_t lds_addr, const float* gptr) {
    uint64_t ga = (uint64_t)(uintptr_t)gptr;
    asm volatile(
        "global_load_async_to_lds_b32 %0, %1, off\n\t"
        "global_load_async_to_lds_b32 %0, %1, off offset:4\n\t"
        "global_load_async_to_lds_b32 %0, %1, off offset:8"
        :: "v"(lds_addr), "v"(ga) : "memory");
}

__device__ __forceinline__ void wait_async_zero() {
    asm volatile("s_wait_asynccnt 0" ::: "memory");
}

__global__ __launch_bounds__(BLOCK) void lapdeform_kernel(
    const float* __restrict__ pcl,
    const float* __restrict__ init_pcl,
    const float* __restrict__ logits,
    const int*   __restrict__ nn,
    float*       __restrict__ out,
    int n)
{
    __shared__ float smem[WAVES_PER_BLOCK * WAVE_LDS_DW];

    const int tid  = (int)threadIdx.x;
    const int lane = tid & 31;
    const int wave = tid >> 5;
    const int i    = (int)blockIdx.x * BLOCK + tid;
    if (i >= n) return;  // n is a multiple of BLOCK; kept for safety

    // ---- neighbor indices (coalesced 4x int4) ----
    const int4* nn4 = (const int4*)(nn + (size_t)i * KNN);
    int4 i0 = nn4[0], i1 = nn4[1], i2 = nn4[2], i3 = nn4[3];
    int idx[KNN] = { i0.x, i0.y, i0.z, i0.w,  i1.x, i1.y, i1.z, i1.w,
                     i2.x, i2.y, i2.z, i2.w,  i3.x, i3.y, i3.z, i3.w };

    // ---- issue all async gathers (2 arrays * 16 neighbors * 12B per lane) ----
    uint32_t lds0    = (uint32_t)(uintptr_t)&smem[wave * WAVE_LDS_DW];
    uint32_t newBase = lds0 + (uint32_t)lane * 12u;
    uint32_t oldBase = newBase + (uint32_t)(OLD_OFF_DW * 4);
#pragma unroll
    for (int k = 0; k < KNN; ++k) {
        const size_t j3 = 3 * (size_t)idx[k];
        async_gather12(newBase + (uint32_t)(k * 32 * 12), pcl + j3);
        async_gather12(oldBase + (uint32_t)(k * 32 * 12), init_pcl + j3);
    }

    // ---- own point data + softmax (overlaps async gather latency) ----
    const float px = pcl[3 * (size_t)i + 0];
    const float py = pcl[3 * (size_t)i + 1];
    const float pz = pcl[3 * (size_t)i + 2];
    const float qx = init_pcl[3 * (size_t)i + 0];
    const float qy = init_pcl[3 * (size_t)i + 1];
    const float qz = init_pcl[3 * (size_t)i + 2];

    const float4* lg4 = (const float4*)(logits + (size_t)i * KNN);
    float4 l0 = lg4[0], l1 = lg4[1], l2 = lg4[2], l3 = lg4[3];
    float l[KNN] = { l0.x, l0.y, l0.z, l0.w,  l1.x, l1.y, l1.z, l1.w,
                     l2.x, l2.y, l2.z, l2.w,  l3.x, l3.y, l3.z, l3.w };
    float m = l[0];
#pragma unroll
    for (int k = 1; k < KNN; ++k) m = fmaxf(m, l[k]);
    float w[KNN];
    float wsum = 0.0f;
#pragma unroll
    for (int k = 0; k < KNN; ++k) { w[k] = __expf(l[k] - m); wsum += w[k]; }
    const float winv = 1.0f / wsum;
#pragma unroll
    for (int k = 0; k < KNN; ++k) w[k] *= winv;

    // ---- wait for gathers, then accumulate from LDS ----
    wait_async_zero();

    const float* snew = &smem[wave * WAVE_LDS_DW + lane * 3];
    const float* sold = snew + OLD_OFF_DW;

    float s00 = 0, s01 = 0, s02 = 0, s10 = 0, s11 = 0, s12 = 0, s20 = 0, s21 = 0, s22 = 0;
    float wnx = 0, wny = 0, wnz = 0;
    bool eqx = true, eqy = true, eqz = true;
#pragma unroll
    for (int k = 0; k < KNN; ++k) {
        const float* pn = snew + k * 96;  // 32 lanes * 3 dwords
        const float* po = sold + k * 96;
        const float nx = pn[0], ny = pn[1], nz = pn[2];
        const float ox = po[0], oy = po[1], oz = po[2];
        const float wk = w[k];
        wnx = fmaf(wk, nx, wnx);
        wny = fmaf(wk, ny, wny);
        wnz = fmaf(wk, nz, wnz);
        const float ex = nx - px, ey = ny - py, ez = nz - pz;   // edges
        const float ux = ox - qx, uy = oy - qy, uz = oz - qz;   // old_edges
        eqx = eqx && (ex == ux);
        eqy = eqy && (ey == uy);
        eqz = eqz && (ez == uz);
        const float a0 = wk * ux, a1 = wk * uy, a2 = wk * uz;
        s00 = fmaf(a0, ex, s00); s01 = fmaf(a0, ey, s01); s02 = fmaf(a0, ez, s02);
        s10 = fmaf(a1, ex, s10); s11 = fmaf(a1, ey, s11); s12 = fmaf(a1, ez, s12);
        s20 = fmaf(a2, ex, s20); s21 = fmaf(a2, ey, s21); s22 = fmaf(a2, ez, s22);
    }

    const float lapx = px - wnx, lapy = py - wny, lapz = pz - wnz;
    const bool unchanged = eqx || eqy || eqz;

    // ---- R = V U^T = orthogonal polar factor of S^T (Higham-Newton) ----
    float a00 = s00, a01 = s10, a02 = s20;   // A = S^T
    float a10 = s01, a11 = s11, a12 = s21;
    float a20 = s02, a21 = s12, a22 = s22;

    const float fro = a00*a00 + a01*a01 + a02*a02 + a10*a10 + a11*a11 + a12*a12
                    + a20*a20 + a21*a21 + a22*a22;
    const bool ident = unchanged || (fro < 1e-24f);
    if (ident) {  // identity in -> identity out of the iteration (branchless-friendly)
        a00 = 1.0f; a01 = 0.0f; a02 = 0.0f;
        a10 = 0.0f; a11 = 1.0f; a12 = 0.0f;
        a20 = 0.0f; a21 = 0.0f; a22 = 1.0f;
    }

#pragma unroll
    for (int it = 0; it < 6; ++it) {
        // cofactor matrix C: A^{-T} = C / det
        const float c00 = a11*a22 - a12*a21;
        const float c01 = a12*a20 - a10*a22;
        const float c02 = a10*a21 - a11*a20;
        const float c10 = a02*a21 - a01*a22;
        const float c11 = a00*a22 - a02*a20;
        const float c12 = a01*a20 - a00*a21;
        const float c20 = a01*a12 - a02*a11;
        const float c21 = a02*a10 - a00*a12;
        const float c22 = a00*a11 - a01*a10;
        const float det = a00*c00 + a01*c01 + a02*c02;
        const float ivd = 1.0f / det;
        // determinant scaling only while far from orthogonality (first 3 steps);
        // afterwards det ~ +-1 and plain Newton converges quadratically.
        float g, h;
        if (it < 3) {
            const float adet = fmaxf(fabsf(det), 1e-30f);
            const float mu   = __powf(adet, -0.3333333333f);  // |det|^(-1/3)
            g = 0.5f * mu;
            h = 0.5f / mu * ivd;
        } else {
            g = 0.5f;
            h = 0.5f * ivd;
        }
        a00 = g*a00 + h*c00;  a01 = g*a01 + h*c01;  a02 = g*a02 + h*c02;
        a10 = g*a10 + h*c10;  a11 = g*a11 + h*c11;  a12 = g*a12 + h*c12;
        a20 = g*a20 + h*c20;  a21 = g*a21 + h*c21;  a22 = g*a22 + h*c22;
    }

    const float r00 = a00, r01 = a01, r02 = a02;
    const float r10 = a10, r11 = a11, r12 = a12;
    const float r20 = a20, r21 = a21, r22 = a22;

    // ---- matrix_to_quaternion (faithful to pytorch3d path) ----
    const float t0 = 1.0f + r00 + r11 + r22;
    const float t1 = 1.0f + r00 - r11 - r22;
    const float t2 = 1.0f - r00 + r11 - r22;
    const float t3 = 1.0f - r00 - r11 + r22;
    const float qa0 = sqrtf(fmaxf(t0, 0.0f));
    const float qa1 = sqrtf(fmaxf(t1, 0.0f));
    const float qa2 = sqrtf(fmaxf(t2, 0.0f));
    const float qa3 = sqrtf(fmaxf(t3, 0.0f));

    int best = 0; float bv = qa0;
    if (qa1 > bv) { bv = qa1; best = 1; }
    if (qa2 > bv) { bv = qa2; best = 2; }
    if (qa3 > bv) { bv = qa3; best = 3; }

    float o0, o1, o2, o3;
    if (best == 0)      { o0 = qa0*qa0;   o1 = r21 - r12; o2 = r02 - r20; o3 = r10 - r01; }
    else if (best == 1) { o0 = r21 - r12; o1 = qa1*qa1;   o2 = r10 + r01; o3 = r02 + r20; }
    else if (best == 2) { o0 = r02 - r20; o1 = r10 + r01; o2 = qa2*qa2;   o3 = r12 + r21; }
    else                { o0 = r10 - r01; o1 = r20 + r02; o2 = r21 + r12; o3 = qa3*qa3;   }
    const float div = 0.5f / fmaxf(bv, 0.1f);

    // ---- coalesced store: stage 224 floats (32 pts x 7) in the wave-private
    // LDS region (gather data is fully consumed; lockstep wave => no barrier),
    // then write contiguous b128s. Wave's out block is 896B => 16B aligned.
    float* stg = &smem[wave * WAVE_LDS_DW];
    stg[lane * 7 + 0] = lapx;
    stg[lane * 7 + 1] = lapy;
    stg[lane * 7 + 2] = lapz;
    stg[lane * 7 + 3] = o0 * div;
    stg[lane * 7 + 4] = o1 * div;
    stg[lane * 7 + 5] = o2 * div;
    stg[lane * 7 + 6] = o3 * div;

    const float4* stg4 = (const float4*)stg;
    float4* out4 = (float4*)(out + ((size_t)blockIdx.x * BLOCK + (size_t)wave * 32) * 7);
    out4[lane] = stg4[lane];                       // float4  0..31
    if (lane < 24) out4[32 + lane] = stg4[32 + lane];  // float4 32..55 (224 floats total)
}

extern "C" void kernel_launch(void* const* d_in, const int* in_sizes, int n_in,
                              void* d_out, int out_size, void* d_ws, size_t ws_size,
                              hipStream_t stream) {
    const float* pcl      = (const float*)d_in[0];
    const float* init_pcl = (const float*)d_in[1];
    const float* logits   = (const float*)d_in[2];
    const int*   nn       = (const int*)d_in[3];
    float*       out      = (float*)d_out;
    const int n = in_sizes[0] / 3;  // N points

    dim3 grid((n + BLOCK - 1) / BLOCK), block(BLOCK);
    hipLaunchKernelGGL(lapdeform_kernel, grid, block, 0, stream,
                       pcl, init_pcl, logits, nn, out, n);
}